// ROIPool_42597485641905
// MI455X (gfx1250) — compile-verified
//
#include <hip/hip_runtime.h>
#include <hip/hip_bf16.h>

// ROI max-pool, MI455X (gfx1250, wave32).
// Output (512,256,7,7) f32 = 25.7 MB -> write-bandwidth bound (~1.1us @ 23.3TB/s).
// Feature map (1,256,50,50) = 2.56 MB -> lives entirely in the 192 MB L2.
// Strategy: 1 thread per output element, flat-linear mapping => perfectly
// coalesced stores; non-temporal store hint for the write-once stream;
// global_prefetch_b8 to warm the window; box math scalarized per block.

#define POOL_P   7
#define POOL_C   256
#define POOL_H   50
#define POOL_W   50
#define NBOXES   512
#define BIN_CNT  (POOL_P * POOL_P)            // 49
#define PER_BOX  (POOL_C * BIN_CNT)           // 12544 (multiple of 32 -> wave aligned)
#define BLOCK    256
#define BLK_PER_BOX (PER_BOX / BLOCK)         // 49

__global__ __launch_bounds__(BLOCK) void ROIPool_42597485641905_kernel(
    const float* __restrict__ feat,     // [256*50*50]
    const float* __restrict__ bboxes,   // [512*4]
    const float* __restrict__ scale_p,  // [1]
    float* __restrict__ out)            // [512*12544]
{
    // box is a pure function of blockIdx -> all box math is SALU / s_load.
    const int box = blockIdx.x / BLK_PER_BOX;
    const int off = (blockIdx.x % BLK_PER_BOX) * BLOCK + threadIdx.x;

    const int c   = off / BIN_CNT;
    const int bin = off - c * BIN_CNT;
    const int ph  = bin / POOL_P;
    const int pw  = bin - ph * POOL_P;

    const float scale = scale_p[0];

    const float4 b = reinterpret_cast<const float4*>(bboxes)[box];
    const int x1 = (int)floorf(b.x * scale + 0.5f);
    const int y1 = (int)floorf(b.y * scale + 0.5f);
    const int x2 = (int)floorf(b.z * scale + 0.5f);
    const int y2 = (int)floorf(b.w * scale + 0.5f);

    const int rw = max(x2 - x1 + 1, 1);
    const int rh = max(y2 - y1 + 1, 1);

    // Bin boundaries (floor start, ceil end), clipped to [0, H] / [0, W].
    int hs = min(max( ph      * rh / POOL_P              + y1, 0), POOL_H);
    int he = min(max(((ph + 1) * rh + POOL_P - 1) / POOL_P + y1, 0), POOL_H);
    int ws = min(max( pw      * rw / POOL_P              + x1, 0), POOL_W);
    int we = min(max(((pw + 1) * rw + POOL_P - 1) / POOL_P + x1, 0), POOL_W);

    float m = 0.0f;  // empty bins yield 0 per the reference
    if (he > hs && we > ws) {
        const float* __restrict__ fc = feat + c * (POOL_H * POOL_W);
        const float* __restrict__ row0 = fc + hs * POOL_W + ws;
        __builtin_prefetch(row0, 0, 0);  // global_prefetch_b8: warm L0/L2

        m = -3.402823466e38f;
        for (int h = hs; h < he; ++h) {
            const float* __restrict__ row = fc + h * POOL_W;
            #pragma clang loop unroll_count(2)
            for (int w = ws; w < we; ++w) {
                m = fmaxf(m, row[w]);
            }
        }
    }

    // Write-once 25.7MB stream: non-temporal hint (TH=NT) keeps it from
    // displacing the L2-resident feature map (and matches the roofline model).
    __builtin_nontemporal_store(m, &out[box * PER_BOX + off]);
}

extern "C" void kernel_launch(void* const* d_in, const int* in_sizes, int n_in,
                              void* d_out, int out_size, void* d_ws, size_t ws_size,
                              hipStream_t stream) {
    const float* feat    = (const float*)d_in[0];  // (1,256,50,50) f32
    const float* bboxes  = (const float*)d_in[1];  // (512,4) f32
    // d_in[2] = output_size (int, ==7, compile-time here)
    const float* scale_p = (const float*)d_in[3];  // (1,) f32
    float* out = (float*)d_out;                    // (512,256,7,7) f32

    const int nblocks = NBOXES * BLK_PER_BOX;      // 25088
    ROIPool_42597485641905_kernel<<<nblocks, BLOCK, 0, stream>>>(feat, bboxes, scale_p, out);
}